// DigitCapsules_38714835206698
// MI455X (gfx1250) — compile-verified
//
#include <hip/hip_runtime.h>

typedef __attribute__((ext_vector_type(2))) float v2f;
typedef __attribute__((ext_vector_type(8))) float v8f;

// Problem constants
#define BATCH   64
#define IN_CAPS 32
#define DSUM    288            // IN_CH * WID * HEI = 8*6*6
#define OUTN    160            // OUT_CAPS * OUT_CH = 10*16
#define OC      10
#define OCH     16
#define USUM_BSTRIDE (IN_CAPS * OUTN)   // 5120 floats per batch row
#define U_BSTRIDE    (IN_CAPS * DSUM)   // 9216 floats per batch row
#define W_ISTRIDE    (DSUM * OUTN)      // 46080 floats per capsule

// ---------------------------------------------------------------------------
// Kernel 1: 32 independent GEMMs  C[64x160] = A[64x288] * B[288x160] via
// V_WMMA_F32_16X16X4_F32 (exact fp32).  One wave per 16x16 output tile.
// grid = 32 (ic) * 4 (M tiles) * 10 (N tiles) = 1280 blocks of 32 threads.
// ---------------------------------------------------------------------------
__global__ __launch_bounds__(32) void usum_wmma(const float* __restrict__ u,
                                                const float* __restrict__ W,
                                                float* __restrict__ us) {
  const int blk = blockIdx.x;
  const int i   = blk / 40;
  const int rem = blk % 40;
  const int m0  = (rem / 10) * 16;   // 0,16,32,48
  const int n0  = (rem % 10) * 16;   // 0..144

  const int l    = threadIdx.x;      // 0..31 (wave32)
  const int half = l >> 4;           // 0: lanes 0-15, 1: lanes 16-31
  const int lm   = l & 15;

  // A[m, k] = u[m*U_BSTRIDE + i*DSUM + k], m = m0+lm (this lane's row)
  const float* __restrict__ Arow = u + (size_t)(m0 + lm) * U_BSTRIDE + (size_t)i * DSUM;
  // B[k, n] = W[i*W_ISTRIDE + k*OUTN + n], n = n0+lm (this lane's column)
  const float* __restrict__ Bcol = W + (size_t)i * W_ISTRIDE + (n0 + lm);

  v8f acc = {0.f, 0.f, 0.f, 0.f, 0.f, 0.f, 0.f, 0.f};

  for (int k0 = 0; k0 < DSUM; k0 += 4) {
    const int ka = k0 + 2 * half;           // lanes 0-15: K=k0,k0+1 ; lanes 16-31: K=k0+2,k0+3
    // A operand: VGPR0 = A[lm, ka], VGPR1 = A[lm, ka+1]  (consecutive -> b64 load)
    const float2 av = *reinterpret_cast<const float2*>(Arow + ka);
    v2f a; a.x = av.x; a.y = av.y;
    // B operand: VGPR0 = B[ka, n0+lm], VGPR1 = B[ka+1, n0+lm]
    v2f b;
    b.x = Bcol[(size_t)ka * OUTN];
    b.y = Bcol[(size_t)(ka + 1) * OUTN];
    acc = __builtin_amdgcn_wmma_f32_16x16x4_f32(false, a, false, b,
                                                (short)0, acc, false, false);
  }

  // C/D layout: VGPR v holds M = m0 + v + 8*half, N = n0 + lm
  float* __restrict__ outp =
      us + (size_t)(m0 + 8 * half) * USUM_BSTRIDE + (size_t)i * OUTN + n0 + lm;
#pragma unroll
  for (int v = 0; v < 8; ++v) outp[(size_t)v * USUM_BSTRIDE] = acc[v];
}

// ---------------------------------------------------------------------------
// Kernel 2: u_dot[i,o] = sum_{b,c} u_sum[b,i,o,c]; init b_ij=0, c_ij=1/32.
// One block, 320 threads (t -> (i = t/10, o = t%10)).
// ---------------------------------------------------------------------------
__global__ __launch_bounds__(320) void init_kernel(const float* __restrict__ us,
                                                   float* __restrict__ udot,
                                                   float* __restrict__ bij,
                                                   float* __restrict__ cij) {
  const int t = threadIdx.x;
  if (t < 320) {
    const int i = t / 10, o = t % 10;
    const float* __restrict__ p = us + (size_t)i * OUTN + o * OCH;
    float s = 0.f;
    for (int b = 0; b < BATCH; ++b) {
      const float* __restrict__ q = p + (size_t)b * USUM_BSTRIDE;
#pragma unroll
      for (int c = 0; c < OCH; ++c) s += q[c];
    }
    udot[t] = s;
    bij[t]  = 0.f;
    cij[t]  = 1.0f / 32.0f;   // softmax of zeros over 32 capsules
  }
}

// ---------------------------------------------------------------------------
// Kernel 3: s_j[b,o,c] = sum_i c_ij[i,o] * u_sum[b,i,o,c].  10240 elements,
// one per thread.  grid = 40 blocks of 256.
// ---------------------------------------------------------------------------
__global__ __launch_bounds__(256) void sj_kernel(const float* __restrict__ us,
                                                 const float* __restrict__ cij,
                                                 float* __restrict__ sj) {
  const int idx = blockIdx.x * 256 + threadIdx.x;   // < 10240
  const int b = idx / OUTN;
  const int n = idx % OUTN;
  const int o = n / OCH;
  const float* __restrict__ p = us + (size_t)b * USUM_BSTRIDE + n;
  float s = 0.f;
#pragma unroll
  for (int i = 0; i < IN_CAPS; ++i) s += cij[i * 10 + o] * p[(size_t)i * OUTN];
  sj[idx] = s;
}

// ---------------------------------------------------------------------------
// Kernel 4: single-block routing update.  Deterministic fixed-order tree
// reductions (no float atomics):
//   n      = sum |s_j|            (global L1 norm)
//   scale  = n / (1 + n^2)        (so v_j = scale * s_j)
//   ssum[o]= sum_{b,c} s_j
//   b_ij  += u_dot[i,o] * scale * ssum[o]
//   c_ij   = softmax(b_ij, axis=i)     (for the next iteration)
// ---------------------------------------------------------------------------
__global__ __launch_bounds__(1024) void route_update(const float* __restrict__ sj,
                                                     const float* __restrict__ udot,
                                                     float* __restrict__ bij,
                                                     float* __restrict__ cij,
                                                     float* __restrict__ scale_out) {
  __shared__ float red[1024];
  __shared__ float ssum[OC];
  __shared__ float scale_sh;
  const int t = threadIdx.x;

  // --- global L1 norm ---
  float a = 0.f;
#pragma unroll
  for (int j = 0; j < 10; ++j) a += fabsf(sj[t + 1024 * j]);
  red[t] = a;
  __syncthreads();
  for (int w = 512; w > 0; w >>= 1) {
    if (t < w) red[t] += red[t + w];
    __syncthreads();
  }
  if (t == 0) {
    const float n = red[0];
    scale_sh = n / (1.0f + n * n);
  }
  __syncthreads();

  // --- per-output-capsule sums: ssum[o] = sum over (b,c) ---
  const int b = t >> 4;        // 0..63
  const int c = t & 15;        // 0..15
  for (int o = 0; o < OC; ++o) {
    red[t] = sj[(size_t)b * OUTN + o * OCH + c];
    __syncthreads();
    for (int w = 512; w > 0; w >>= 1) {
      if (t < w) red[t] += red[t + w];
      __syncthreads();
    }
    if (t == 0) ssum[o] = red[0];
    __syncthreads();
  }

  // --- b_ij update ---
  if (t < 320) {
    const int o = t % 10;
    bij[t] += udot[t] * (scale_sh * ssum[o]);
  }
  __syncthreads();

  // --- softmax over i (32) for each column o: 10 threads, serial columns ---
  if (t < OC) {
    const int o = t;
    float mx = -3.0e38f;
    for (int i = 0; i < IN_CAPS; ++i) mx = fmaxf(mx, bij[i * 10 + o]);
    float sum = 0.f;
    for (int i = 0; i < IN_CAPS; ++i) sum += expf(bij[i * 10 + o] - mx);
    const float inv = 1.0f / sum;
    for (int i = 0; i < IN_CAPS; ++i) cij[i * 10 + o] = expf(bij[i * 10 + o] - mx) * inv;
  }
  if (t == 0) *scale_out = scale_sh;
}

// ---------------------------------------------------------------------------
// Kernel 5: v_j = scale * s_j  -> d_out
// ---------------------------------------------------------------------------
__global__ __launch_bounds__(256) void finalize(const float* __restrict__ sj,
                                                const float* __restrict__ scale,
                                                float* __restrict__ out) {
  const int idx = blockIdx.x * 256 + threadIdx.x;
  if (idx < BATCH * OUTN) out[idx] = (*scale) * sj[idx];
}

// ---------------------------------------------------------------------------
extern "C" void kernel_launch(void* const* d_in, const int* in_sizes, int n_in,
                              void* d_out, int out_size, void* d_ws, size_t ws_size,
                              hipStream_t stream) {
  const float* u = (const float*)d_in[0];   // [64,32,8,6,6]
  const float* W = (const float*)d_in[1];   // [32,8,6,6,10,16]
  float* out = (float*)d_out;               // [64,10,16]

  float* ws    = (float*)d_ws;
  float* us    = ws;                        // 327680 floats: u_sum [64,32,10,16]
  float* udot  = us + (size_t)BATCH * USUM_BSTRIDE;   // 320
  float* bij   = udot + 320;                // 320
  float* cij   = bij + 320;                 // 320
  float* sj    = cij + 320;                 // 10240
  float* scale = sj + BATCH * OUTN;         // 1

  // Main GEMM batch: 32 x (64x288 @ 288x160) via fp32 WMMA
  usum_wmma<<<IN_CAPS * 4 * OC, 32, 0, stream>>>(u, W, us);

  // u_dot + routing state init
  init_kernel<<<1, 320, 0, stream>>>(us, udot, bij, cij);

  // 3 routing iterations (sequential due to global L1 norm)
  for (int it = 0; it < 3; ++it) {
    sj_kernel<<<40, 256, 0, stream>>>(us, cij, sj);
    route_update<<<1, 1024, 0, stream>>>(sj, udot, bij, cij, scale);
  }

  // v_j from the final iteration
  finalize<<<40, 256, 0, stream>>>(sj, scale, out);
}